// SpatialROIExtractorEnhanced_4818953306561
// MI455X (gfx1250) — compile-verified
//
#include <hip/hip_runtime.h>
#include <hip/hip_bf16.h>

typedef __attribute__((ext_vector_type(2))) float v2f;
typedef __attribute__((ext_vector_type(8))) float v8f;

// ---------------------------------------------------------------------------
// Kernel 1: ROI align (torchvision semantics, aligned=true, adaptive sampling)
// one block = one box; separable sample tables in LDS, then C*49 outputs.
// ---------------------------------------------------------------------------
__global__ __launch_bounds__(256) void roi_align_kernel(
    const float* __restrict__ f,      // [128,120,120]
    const float* __restrict__ boxes,  // [512,4] normalized xywh
    float* __restrict__ flat)         // [512, 128*49]
{
    constexpr int C = 128, H = 120, W = 120;
    constexpr int R = 7, SMAX = 5;

    const int box = blockIdx.x;
    const int tid = threadIdx.x;

    __shared__ int   y0s[R * SMAX], y1s[R * SMAX], x0s[R * SMAX], x1s[R * SMAX];
    __shared__ float wy0s[R * SMAX], wy1s[R * SMAX], wx0s[R * SMAX], wx1s[R * SMAX];
    __shared__ int   s_gh, s_gw;
    __shared__ float s_invcnt;

    const float bx = boxes[box * 4 + 0];
    const float by = boxes[box * 4 + 1];
    const float bw = boxes[box * 4 + 2];
    const float bh = boxes[box * 4 + 3];

    const float x1 = bx * (float)W - 0.5f;
    const float y1 = by * (float)H - 0.5f;
    const float roi_w = bw * (float)W;
    const float roi_h = bh * (float)H;
    const float bin_w = roi_w / (float)R;
    const float bin_h = roi_h / (float)R;
    const float ghf = fmaxf(ceilf(roi_h / (float)R), 1.0f);
    const float gwf = fmaxf(ceilf(roi_w / (float)R), 1.0f);

    if (tid < 2 * R * SMAX) {
        const bool isY  = tid < R * SMAX;
        const int  e    = isY ? tid : tid - R * SMAX;
        const int  ph   = e / SMAX;
        const int  s    = e - ph * SMAX;
        const float g     = isY ? ghf : gwf;
        const float start = isY ? y1 : x1;
        const float bin   = isY ? bin_h : bin_w;
        const float lim   = isY ? (float)H : (float)W;

        const float coord = start + (float)ph * bin + ((float)s + 0.5f) * bin / g;
        const bool valid  = (coord > -1.0f) && (coord < lim) && ((float)s < g);
        const float cc = fminf(fmaxf(coord, 0.0f), lim - 1.0f);
        const int  c0  = (int)floorf(cc);
        const int  c1  = min(c0 + 1, (int)lim - 1);
        const float l  = cc - (float)c0;
        const float whi = valid ? (1.0f - l) : 0.0f;  // weight for floor neighbor
        const float wlo = valid ? l          : 0.0f;  // weight for ceil neighbor
        if (isY) { y0s[e] = c0; y1s[e] = c1; wy0s[e] = whi; wy1s[e] = wlo; }
        else     { x0s[e] = c0; x1s[e] = c1; wx0s[e] = whi; wx1s[e] = wlo; }
    }
    if (tid == 0) {
        s_gh = (int)ghf;
        s_gw = (int)gwf;
        s_invcnt = 1.0f / (ghf * gwf);
    }
    __syncthreads();

    const int gh = s_gh, gw = s_gw;
    const float invc = s_invcnt;

    for (int o = tid; o < C * R * R; o += 256) {
        const int c   = o / (R * R);
        const int bin = o - c * (R * R);
        const int ph  = bin / R;
        const int pw  = bin - ph * R;
        const float* fc = f + (size_t)c * (H * W);

        float acc = 0.0f;
        for (int sy = 0; sy < gh; ++sy) {
            const int   ey  = ph * SMAX + sy;
            const int   ya  = y0s[ey] * W;
            const int   yb  = y1s[ey] * W;
            const float wy0 = wy0s[ey];
            const float wy1 = wy1s[ey];
            for (int sx = 0; sx < gw; ++sx) {
                const int   ex  = pw * SMAX + sx;
                const int   xa  = x0s[ex];
                const int   xb  = x1s[ex];
                const float wx0 = wx0s[ex];
                const float wx1 = wx1s[ex];
                acc += wy0 * (wx0 * fc[ya + xa] + wx1 * fc[ya + xb])
                     + wy1 * (wx0 * fc[yb + xa] + wx1 * fc[yb + xb]);
            }
        }
        flat[(size_t)box * (C * R * R) + o] = acc * invc;
    }
}

// ---------------------------------------------------------------------------
// Kernel 2/3: software-pipelined tiled fp32 GEMM on v_wmma_f32_16x16x4_f32.
// Block: 256 threads = 8 waves, tile BM=64 x BN=32 (waves 4x2 of 16x16).
// A [M,K], B [K,N] row-major; C = act(A*B + bias).
//
// Pipeline: next K-block's A/B tiles are prefetched into registers BEFORE the
// WMMA chain of the current tile, so global latency overlaps compute; the
// regs are spilled to LDS after the post-compute barrier.
//
// B tile is stored TRANSPOSED in LDS (Bt[n][k]) so both A and B fragments are
// single 8-byte ds loads (two consecutive K values at fixed row/col).
//
// Fragment layouts per CDNA5 ISA 7.12.2 (wave32):
//   A 16x4 : VGPR0/1 = K0,K1 (lanes 0-15) | K2,K3 (lanes 16-31), row M = lane%16
//   B 4x16 : VGPRr   = row K=r (lanes 0-15) | row K=r+2 (lanes 16-31), col N = lane%16
//   C 16x16: VGPRr   = M=r (lanes 0-15) | M=r+8 (lanes 16-31), col N = lane%16
// ---------------------------------------------------------------------------
template <bool RELU>
__global__ __launch_bounds__(256) void wmma_gemm_f32(
    const float* __restrict__ A, const float* __restrict__ B,
    const float* __restrict__ bias, float* __restrict__ C,
    int M, int N, int K)
{
    constexpr int BM = 64, BN = 32, KC = 32;
    constexpr int AS = KC + 2;  // 34 (even -> 8B-aligned v2f rows in LDS)
    constexpr int BS = KC + 2;  // Bt row stride, also even

    __shared__ __align__(16) float As[BM * AS];       // As[m][k]
    __shared__ __align__(16) float Bt[BN * BS];       // Bt[n][k]  (transposed)

    const int tid  = threadIdx.x;
    const int lane = tid & 31;
    const int wave = tid >> 5;     // 0..7
    const int wm   = wave >> 1;    // 0..3  (M sub-tile)
    const int wn   = wave & 1;     // 0..1  (N sub-tile)
    const int half = lane >> 4;    // 0|1
    const int l16  = lane & 15;

    const int m0 = blockIdx.y * BM;
    const int n0 = blockIdx.x * BN;

    // Static staging coordinates for this thread
    const int arow0 = tid >> 3;               // A rows 0..31   (8 float4 per row)
    const int arow1 = (tid + 256) >> 3;       // A rows 32..63
    const int ac4   = (tid & 7) << 2;         // k offset within KC
    const int brow  = tid >> 3;               // k within KC block
    const int bc4   = (tid & 7) << 2;         // n offset within BN

    v8f acc = {};

    // ---- prologue: fetch first tiles into registers ----
    float4 ra0 = *reinterpret_cast<const float4*>(A + (size_t)(m0 + arow0) * K + ac4);
    float4 ra1 = *reinterpret_cast<const float4*>(A + (size_t)(m0 + arow1) * K + ac4);
    float4 rb  = *reinterpret_cast<const float4*>(B + (size_t)brow * N + n0 + bc4);

    for (int kb = 0; kb < K; kb += KC) {
        // ---- spill staged registers to LDS ----
        {
            float* d0 = &As[arow0 * AS + ac4];
            d0[0] = ra0.x; d0[1] = ra0.y; d0[2] = ra0.z; d0[3] = ra0.w;
            float* d1 = &As[arow1 * AS + ac4];
            d1[0] = ra1.x; d1[1] = ra1.y; d1[2] = ra1.z; d1[3] = ra1.w;
            // transposed scatter: B[kb+brow][n0+bc4+j] -> Bt[bc4+j][brow]
            Bt[(bc4 + 0) * BS + brow] = rb.x;
            Bt[(bc4 + 1) * BS + brow] = rb.y;
            Bt[(bc4 + 2) * BS + brow] = rb.z;
            Bt[(bc4 + 3) * BS + brow] = rb.w;
        }
        __syncthreads();

        // ---- prefetch next tiles into registers (overlaps WMMA chain) ----
        const int kn = kb + KC;
        if (kn < K) {
            ra0 = *reinterpret_cast<const float4*>(A + (size_t)(m0 + arow0) * K + kn + ac4);
            ra1 = *reinterpret_cast<const float4*>(A + (size_t)(m0 + arow1) * K + kn + ac4);
            rb  = *reinterpret_cast<const float4*>(B + (size_t)(kn + brow) * N + n0 + bc4);
        }

        // ---- WMMA chain over current LDS tiles ----
        const float* aRow = &As[(wm * 16 + l16) * AS];
        const float* bRow = &Bt[(wn * 16 + l16) * BS];
        #pragma unroll
        for (int k = 0; k < KC; k += 4) {
            const int kk = k + half * 2;
            v2f a = *reinterpret_cast<const v2f*>(aRow + kk);
            v2f b = *reinterpret_cast<const v2f*>(bRow + kk);
            acc = __builtin_amdgcn_wmma_f32_16x16x4_f32(
                false, a, false, b, (short)0, acc, false, false);
        }
        __syncthreads();
    }

    // ---- epilogue: bias + optional ReLU, scatter per C/D layout ----
    const int col = n0 + wn * 16 + l16;
    const float bv = bias[col];
    #pragma unroll
    for (int r = 0; r < 8; ++r) {
        const int row = m0 + wm * 16 + r + half * 8;
        float v = acc[r] + bv;
        if (RELU) v = fmaxf(v, 0.0f);
        C[(size_t)row * N + col] = v;
    }
}

// ---------------------------------------------------------------------------
// Launcher
// ---------------------------------------------------------------------------
extern "C" void kernel_launch(void* const* d_in, const int* in_sizes, int n_in,
                              void* d_out, int out_size, void* d_ws, size_t ws_size,
                              hipStream_t stream) {
    (void)in_sizes; (void)n_in; (void)out_size; (void)ws_size;

    const float* features = (const float*)d_in[0];  // [1,128,120,120]
    const float* boxes    = (const float*)d_in[1];  // [512,4]
    const float* W1       = (const float*)d_in[2];  // [6272,512]
    const float* b1       = (const float*)d_in[3];  // [512]
    const float* W2       = (const float*)d_in[4];  // [512,256]
    const float* b2       = (const float*)d_in[5];  // [256]
    float*       out      = (float*)d_out;          // [512,256]

    constexpr int NBOX = 512;
    constexpr int DIN  = 128 * 7 * 7;   // 6272
    constexpr int DH   = 512;
    constexpr int DOUT = 256;

    float* flat = (float*)d_ws;                      // [512, 6272]
    float* Hbuf = flat + (size_t)NBOX * DIN;         // [512, 512]

    // 1) ROI align -> flat
    roi_align_kernel<<<NBOX, 256, 0, stream>>>(features, boxes, flat);

    // 2) H = relu(flat @ W1 + b1) : M=512, N=512, K=6272
    wmma_gemm_f32<true><<<dim3(DH / 32, NBOX / 64), 256, 0, stream>>>(
        flat, W1, b1, Hbuf, NBOX, DH, DIN);

    // 3) out = H @ W2 + b2 : M=512, N=256, K=512
    wmma_gemm_f32<false><<<dim3(DOUT / 32, NBOX / 64), 256, 0, stream>>>(
        Hbuf, W2, b2, out, NBOX, DOUT, DH);
}